// SpecialSpmmFinal_16612933501031
// MI455X (gfx1250) — compile-verified
//
#include <hip/hip_runtime.h>
#include <stdint.h>

// ---------------------------------------------------------------------------
// segment-sum scatter: out[src[e], 0:16] += edge_w[e, 0:16]
// Memory-bound (~224 MB @ 23.3 TB/s => ~10us floor). Strategy:
//   - TDM (tensor_load_to_lds) streams contiguous edge_w/index tiles to LDS,
//     double-buffered, tracked with TENSORcnt (gfx1250 async-tensor path).
//   - f32 global atomics (no-return) drain into L2 (out = 6.4MB << 192MB L2).
// ---------------------------------------------------------------------------

typedef __attribute__((ext_vector_type(4))) unsigned int v4u;
typedef __attribute__((ext_vector_type(8))) int          v8i;
typedef __attribute__((ext_vector_type(4))) int          v4i;

#define TILE_EDGES      256      // 256 edges * 64B = 16KB edge_w tile, 1KB idx tile
#define TILES_PER_BLOCK 4
#define BLOCK_THREADS   256      // 8 waves (wave32)

// 1-D contiguous TDM copy: tile_units elements of (1<<ds_log) bytes each,
// global OOB clamped via tensor_dim0 = tensor_units (reads past end -> 0).
__device__ __forceinline__ void tdm_load_1d(uint32_t lds_off, const void* gptr,
                                            uint32_t ds_log, uint32_t tile_units,
                                            uint32_t tensor_units) {
  uint64_t ga = (uint64_t)(uintptr_t)gptr;
  v4u g0;
  g0.x = 1u;                                                   // count=1, user mode
  g0.y = lds_off;                                              // lds_addr (bytes)
  g0.z = (uint32_t)ga;                                         // global_addr[31:0]
  g0.w = ((uint32_t)(ga >> 32) & 0x01FFFFFFu) | (2u << 30);    // addr[56:32] | type=2

  v8i g1;
  g1[0] = (int)(ds_log << 16);                                 // mask=0 | data_size
  g1[1] = (int)((tensor_units & 0xFFFFu) << 16);               // tensor_dim0[15:0]
  g1[2] = (int)(((tensor_units >> 16) & 0xFFFFu) | (1u << 16));// dim0[31:16] | tensor_dim1=1
  g1[3] = (int)((tile_units & 0xFFFFu) << 16);                 // tensor_dim1 hi | tile_dim0
  g1[4] = 1;                                                   // tile_dim1=1, tile_dim2=0
  g1[5] = (int)tile_units;                                     // tensor_dim0_stride[31:0]
  g1[6] = (int)((tile_units & 0xFFFFu) << 16);                 // stride0 hi=0 | stride1 lo
  g1[7] = (int)(tile_units >> 16);                             // stride1 hi
  v4i gz4 = {0, 0, 0, 0};
  v8i gz8 = {0, 0, 0, 0, 0, 0, 0, 0};
  __builtin_amdgcn_tensor_load_to_lds(g0, g1, gz4, gz4, gz8, 0);
}

__global__ void zero_out_kernel(float4* __restrict__ o, int n4) {
  int i = blockIdx.x * blockDim.x + threadIdx.x;
  if (i < n4) o[i] = float4{0.f, 0.f, 0.f, 0.f};
}

__global__ __launch_bounds__(BLOCK_THREADS)
void seg_scatter_kernel(const int* __restrict__ src, const float* __restrict__ ew,
                        float* __restrict__ out, int E, int numTiles) {
  __shared__ float s_w[2][TILE_EDGES * 16];   // 2 x 16KB
  __shared__ int   s_i[2][TILE_EDGES];        // 2 x 1KB

  const int tile0 = blockIdx.x * TILES_PER_BLOCK;
  int tEnd = tile0 + TILES_PER_BLOCK;
  if (tEnd > numTiles) tEnd = numTiles;
  const int nt = tEnd - tile0;
  if (nt <= 0) return;

  const bool issuer = (threadIdx.x < 32);     // wave 0 drives the TDM pipeline

  auto issue = [&](int t, int b) {
    const int eb = (tile0 + t) * TILE_EDGES;
    const uint32_t remW = (uint32_t)(E - eb) * 8u;   // 8B units left in edge_w
    const uint32_t remI = (uint32_t)(E - eb);        // 4B units left in src
    tdm_load_1d((uint32_t)(uintptr_t)&s_w[b][0], ew + (size_t)eb * 16,
                /*8B*/ 3, TILE_EDGES * 8, remW);
    tdm_load_1d((uint32_t)(uintptr_t)&s_i[b][0], src + eb,
                /*4B*/ 2, TILE_EDGES, remI);
  };

  if (issuer) issue(0, 0);

  for (int t = 0; t < nt; ++t) {
    if (issuer) {
      if (t + 1 < nt) {
        issue(t + 1, (t + 1) & 1);                  // prefetch next tile
        __builtin_amdgcn_s_wait_tensorcnt(2);       // current tile's 2 TDM ops done
      } else {
        __builtin_amdgcn_s_wait_tensorcnt(0);
      }
    }
    __syncthreads();                                // publish LDS tile to all waves

    const float* w  = s_w[t & 1];
    const int*   si = s_i[t & 1];
    const int    eb = (tile0 + t) * TILE_EDGES;

    // 4 lanes per edge; each lane: one float4 LDS read + 4 f32 global atomics.
    for (int j = threadIdx.x; j < TILE_EDGES * 4; j += BLOCK_THREADS) {
      const int el = j >> 2;
      const int c  = j & 3;
      const int e  = eb + el;
      if (e < E) {
        const int node = si[el];
        const float4 v = *reinterpret_cast<const float4*>(&w[el * 16 + c * 4]);
        float* dst = out + (size_t)node * 16 + c * 4;
        (void)__hip_atomic_fetch_add(dst + 0, v.x, __ATOMIC_RELAXED, __HIP_MEMORY_SCOPE_AGENT);
        (void)__hip_atomic_fetch_add(dst + 1, v.y, __ATOMIC_RELAXED, __HIP_MEMORY_SCOPE_AGENT);
        (void)__hip_atomic_fetch_add(dst + 2, v.z, __ATOMIC_RELAXED, __HIP_MEMORY_SCOPE_AGENT);
        (void)__hip_atomic_fetch_add(dst + 3, v.w, __ATOMIC_RELAXED, __HIP_MEMORY_SCOPE_AGENT);
      }
    }
    __syncthreads();                                // buffer free for next prefetch
  }
}

extern "C" void kernel_launch(void* const* d_in, const int* in_sizes, int n_in,
                              void* d_out, int out_size, void* d_ws, size_t ws_size,
                              hipStream_t stream) {
  const int*   edge = (const int*)d_in[0];   // [2, E]; row 0 = src
  const float* ew   = (const float*)d_in[1]; // [E, 16]
  float*       out  = (float*)d_out;         // [N, 16]
  const int E = in_sizes[1] / 16;

  // out is poisoned by the harness -> zero it first (out_size = N*16, mult of 4).
  const int n4 = out_size / 4;
  zero_out_kernel<<<(n4 + 255) / 256, 256, 0, stream>>>((float4*)out, n4);

  const int numTiles = (E + TILE_EDGES - 1) / TILE_EDGES;
  const int grid = (numTiles + TILES_PER_BLOCK - 1) / TILES_PER_BLOCK;
  seg_scatter_kernel<<<grid, BLOCK_THREADS, 0, stream>>>(edge, ew, out, E, numTiles);
}